// PointNetLK_3444563771750
// MI455X (gfx1250) — compile-verified
//
#include <hip/hip_runtime.h>
#include <math.h>

typedef __attribute__((ext_vector_type(2))) float v2f;
typedef __attribute__((ext_vector_type(8))) float v8f;

#define XTOL_F 1e-7f
#define B_BATCH 16
#define NPTS 1024

// ---------------------------------------------------------------------------
// exp_se3: 4x4 row-major from 6-twist (w | v), matches the JAX reference.
// ---------------------------------------------------------------------------
__device__ __forceinline__ void exp_se3_dev(const float x[6], float T[16]) {
  const float wx = x[0], wy = x[1], wz = x[2];
  const float vx = x[3], vy = x[4], vz = x[5];
  const float t2 = wx*wx + wy*wy + wz*wz;
  const bool small = t2 < 1e-12f;
  const float t2s = small ? 1.f : t2;
  const float t   = sqrtf(t2s);
  const float A  = small ? (1.f - t2 * (1.f/6.f))           : (sinf(t) / t);
  const float Bc = small ? (0.5f - t2 * (1.f/24.f))         : ((1.f - cosf(t)) / t2s);
  const float Cc = small ? ((1.f/6.f) - t2 * (1.f/120.f))   : ((t - sinf(t)) / (t2s * t));
  const float W[9] = {0.f, -wz,  wy,
                      wz,  0.f, -wx,
                     -wy,  wx,  0.f};
  float W2[9];
  #pragma unroll
  for (int i = 0; i < 3; ++i)
    #pragma unroll
    for (int j = 0; j < 3; ++j) {
      float s = 0.f;
      #pragma unroll
      for (int k = 0; k < 3; ++k) s += W[i*3+k] * W[k*3+j];
      W2[i*3+j] = s;
    }
  float R[9], V[9];
  #pragma unroll
  for (int i = 0; i < 9; ++i) {
    const float I = (i % 4 == 0) ? 1.f : 0.f;
    R[i] = I + A  * W[i] + Bc * W2[i];
    V[i] = I + Bc * W[i] + Cc * W2[i];
  }
  const float tvx = V[0]*vx + V[1]*vy + V[2]*vz;
  const float tvy = V[3]*vx + V[4]*vy + V[5]*vz;
  const float tvz = V[6]*vx + V[7]*vy + V[8]*vz;
  T[0]=R[0]; T[1]=R[1]; T[2] =R[2]; T[3] =tvx;
  T[4]=R[3]; T[5]=R[4]; T[6] =R[5]; T[7] =tvy;
  T[8]=R[6]; T[9]=R[7]; T[10]=R[8]; T[11]=tvz;
  T[12]=0.f; T[13]=0.f; T[14]=0.f;  T[15]=1.f;
}

// ---------------------------------------------------------------------------
// Centering: mean over 1024 points + centered copies. 32 blocks (16 batches x
// {template, source}).
// ---------------------------------------------------------------------------
__global__ __launch_bounds__(256)
void k_center(const float* __restrict__ tpl, const float* __restrict__ src,
              float* __restrict__ m0, float* __restrict__ m1,
              float* __restrict__ t0, float* __restrict__ s0) {
  const int b = blockIdx.x & 15;
  const int which = blockIdx.x >> 4;
  const float* in  = which ? src : tpl;
  float*       mean = which ? m1 : m0;
  float*       out  = which ? s0 : t0;
  __shared__ float sx[256], sy[256], sz[256];
  const int tid = threadIdx.x;
  float ax = 0.f, ay = 0.f, az = 0.f;
  for (int n = tid; n < NPTS; n += 256) {
    const float* q = in + (b*NPTS + n)*3;
    ax += q[0]; ay += q[1]; az += q[2];
  }
  sx[tid] = ax; sy[tid] = ay; sz[tid] = az;
  __syncthreads();
  for (int s = 128; s > 0; s >>= 1) {
    if (tid < s) { sx[tid] += sx[tid+s]; sy[tid] += sy[tid+s]; sz[tid] += sz[tid+s]; }
    __syncthreads();
  }
  if (tid == 0) {
    sx[0] *= (1.f/NPTS); sy[0] *= (1.f/NPTS); sz[0] *= (1.f/NPTS);
    mean[b*3+0] = sx[0]; mean[b*3+1] = sy[0]; mean[b*3+2] = sz[0];
  }
  __syncthreads();
  const float mx = sx[0], my = sy[0], mz = sz[0];
  for (int n = tid; n < NPTS; n += 256) {
    const float* q = in  + (b*NPTS + n)*3;
    float*       o = out + (b*NPTS + n)*3;
    o[0] = q[0]-mx; o[1] = q[1]-my; o[2] = q[2]-mz;
  }
}

// ---------------------------------------------------------------------------
// Build the 96 finite-difference point sets: exp_se3(-dt[k] * e_k) applied to t0.
// ---------------------------------------------------------------------------
__global__ __launch_bounds__(256)
void k_jacpts(const float* __restrict__ t0, const float* __restrict__ dt,
              float* __restrict__ jp) {
  const int blk = blockIdx.x;          // b*6 + k
  const int b = blk / 6, k = blk % 6;
  float x6[6] = {0.f,0.f,0.f,0.f,0.f,0.f};
  x6[k] = -dt[k];
  float T[16];
  exp_se3_dev(x6, T);
  const int tid = threadIdx.x;
  for (int n = tid; n < NPTS; n += 256) {
    const float* q = t0 + (b*NPTS + n)*3;
    const float X = q[0], Y = q[1], Z = q[2];
    float* o = jp + (blk*NPTS + n)*3;
    o[0] = T[0]*X + T[1]*Y + T[2] *Z + T[3];
    o[1] = T[4]*X + T[5]*Y + T[6] *Z + T[7];
    o[2] = T[8]*X + T[9]*Y + T[10]*Z + T[11];
  }
}

// ---------------------------------------------------------------------------
// Small MLP layers (16-point tile, 256 threads, weights via L2).
// ---------------------------------------------------------------------------
template<int CI, int CO>
__device__ __forceinline__ void layerMLP(const float* __restrict__ W,
                                         const float* __restrict__ bias,
                                         const float* __restrict__ in,
                                         float* __restrict__ out, int tid) {
  constexpr int GROUPS = 256 / CO;     // 4 (CO=64) or 2 (CO=128)
  constexpr int PTS    = 16 / GROUPS;  // 4 or 8 points per thread
  const int co = tid % CO;
  const int g  = tid / CO;
  float acc[PTS];
  const float bv = bias[co];
  #pragma unroll
  for (int p = 0; p < PTS; ++p) acc[p] = bv;
  for (int k = 0; k < CI; ++k) {
    const float w = W[k*CO + co];
    #pragma unroll
    for (int p = 0; p < PTS; ++p)
      acc[p] = fmaf(in[(g*PTS + p)*CI + k], w, acc[p]);
  }
  #pragma unroll
  for (int p = 0; p < PTS; ++p) {
    const float v = acc[p];
    out[(g*PTS + p)*CO + co] = v > 0.f ? v : 0.f;  // ReLU
  }
}

// ---------------------------------------------------------------------------
// Feature extractor: one workgroup (8 wave32) per point set.
// Layers 1-4: VALU via LDS. Layer 5 (128->1024, 89% of FLOPs):
// fp32 WMMA v_wmma_f32_16x16x4_f32, 16pts x 16ch tiles, K=128 chained.
// Then bias + ReLU + cross-lane (xor 16) max-pool, running max over 64 tiles.
// ---------------------------------------------------------------------------
__global__ __launch_bounds__(256)
void k_feat(const float* __restrict__ pts,
            const float* __restrict__ W1, const float* __restrict__ b1,
            const float* __restrict__ W2, const float* __restrict__ b2,
            const float* __restrict__ W3, const float* __restrict__ b3,
            const float* __restrict__ W4, const float* __restrict__ b4,
            const float* __restrict__ W5, const float* __restrict__ b5,
            float* __restrict__ feat,
            const int* maxiter, int iter) {
  if (maxiter && iter >= *maxiter) return;   // uniform: safe for WMMA EXEC
  __shared__ float sA[16*64];   // layer1 / layer3 outputs
  __shared__ float sB[16*128];  // layer2 / layer4 outputs (WMMA A source)
  __shared__ float sP[48];      // 16 points x 3
  const int tid   = threadIdx.x;
  const int set   = blockIdx.x;
  const int lane  = tid & 31;
  const int wave  = tid >> 5;
  const int laneM = lane & 15;
  const int koff  = (lane >> 4) << 1;   // 0 for lanes 0-15, 2 for lanes 16-31

  float maxv[8];
  #pragma unroll
  for (int i = 0; i < 8; ++i) maxv[i] = 0.f;  // ReLU outputs are >= 0

  for (int ptile = 0; ptile < 64; ++ptile) {
    if (tid < 48) sP[tid] = pts[set*NPTS*3 + ptile*48 + tid];
    __syncthreads();
    layerMLP<3, 64 >(W1, b1, sP, sA, tid); __syncthreads();
    layerMLP<64,64 >(W2, b2, sA, sB, tid); __syncthreads();
    layerMLP<64,64 >(W3, b3, sB, sA, tid); __syncthreads();
    layerMLP<64,128>(W4, b4, sA, sB, tid); __syncthreads();

    // A fragment source: acts[pt][k], pt = laneM, k-pair selected by koff.
    const float* arow = &sB[laneM*128 + koff];
    #pragma unroll
    for (int nt = 0; nt < 8; ++nt) {
      const int n0 = (wave*8 + nt) * 16;
      const float* wcol = W5 + koff*1024 + n0 + laneM;   // W5[k][n], row-major 128x1024
      __builtin_prefetch(wcol + 64*1024, 0, 1);          // -> global_prefetch
      v8f acc = {};
      #pragma unroll 4
      for (int ks = 0; ks < 32; ++ks) {                  // K = 128 in steps of 4
        v2f a;  a.x  = arow[ks*4];        a.y  = arow[ks*4 + 1];
        v2f bm; bm.x = wcol[ks*4096];     bm.y = wcol[ks*4096 + 1024];
        acc = __builtin_amdgcn_wmma_f32_16x16x4_f32(
                  /*neg_a=*/false, a, /*neg_b=*/false, bm,
                  /*c_mod=*/(short)0, acc, /*reuse_a=*/false, /*reuse_b=*/false);
      }
      const float bias = b5[n0 + laneM];
      float m = 0.f;
      #pragma unroll
      for (int r = 0; r < 8; ++r) {       // max over M = r (+8 on upper half-lanes)
        float v = acc[r] + bias;
        v = v > 0.f ? v : 0.f;
        m = fmaxf(m, v);
      }
      m = fmaxf(m, __shfl_xor(m, 16, 32)); // combine M=0..7 with M=8..15 (wave32)
      maxv[nt] = fmaxf(maxv[nt], m);
    }
    __syncthreads();
  }
  if (lane < 16) {
    #pragma unroll
    for (int nt = 0; nt < 8; ++nt)
      feat[set*1024 + (wave*8 + nt)*16 + laneM] = maxv[nt];
  }
}

// ---------------------------------------------------------------------------
// J from finite differences, H = J^T J (ds_add_f32 LDS atomics), 6x6
// Gauss-Jordan inverse, pinv = H^-1 J^T. One block per batch.
// ---------------------------------------------------------------------------
__global__ __launch_bounds__(256)
void k_pinv(const float* __restrict__ tf, const float* __restrict__ fjac,
            const float* __restrict__ dt, float* __restrict__ pinvM) {
  const int b = blockIdx.x;
  const int tid = threadIdx.x;
  __shared__ float sH[36];
  __shared__ float sHinv[36];
  if (tid < 36) sH[tid] = 0.f;
  __syncthreads();
  float invdt[6];
  #pragma unroll
  for (int k = 0; k < 6; ++k) invdt[k] = 1.f / dt[k];

  float h[21];
  #pragma unroll
  for (int i = 0; i < 21; ++i) h[i] = 0.f;
  for (int n = tid; n < NPTS; n += 256) {
    const float tfn = tf[b*1024 + n];
    float j6[6];
    #pragma unroll
    for (int k = 0; k < 6; ++k)
      j6[k] = (tfn - fjac[(b*6 + k)*1024 + n]) * invdt[k];
    int idx = 0;
    #pragma unroll
    for (int i2 = 0; i2 < 6; ++i2)
      #pragma unroll
      for (int j2 = i2; j2 < 6; ++j2) h[idx++] += j6[i2] * j6[j2];
  }
  {
    int idx = 0;
    #pragma unroll
    for (int i2 = 0; i2 < 6; ++i2)
      #pragma unroll
      for (int j2 = i2; j2 < 6; ++j2) atomicAdd(&sH[i2*6 + j2], h[idx++]);
  }
  __syncthreads();
  if (tid == 0) {
    float M[6][12];
    for (int i = 0; i < 6; ++i)
      for (int j = 0; j < 6; ++j) {
        M[i][j]   = (j >= i) ? sH[i*6 + j] : sH[j*6 + i];
        M[i][6+j] = (i == j) ? 1.f : 0.f;
      }
    for (int c = 0; c < 6; ++c) {       // H is SPD: no pivoting needed
      const float ip = 1.f / M[c][c];
      for (int j = 0; j < 12; ++j) M[c][j] *= ip;
      for (int r = 0; r < 6; ++r) if (r != c) {
        const float f = M[r][c];
        for (int j = 0; j < 12; ++j) M[r][j] -= f * M[c][j];
      }
    }
    for (int i = 0; i < 6; ++i)
      for (int j = 0; j < 6; ++j) sHinv[i*6 + j] = M[i][6+j];
  }
  __syncthreads();
  for (int n = tid; n < NPTS; n += 256) {
    const float tfn = tf[b*1024 + n];
    float j6[6];
    #pragma unroll
    for (int k = 0; k < 6; ++k)
      j6[k] = (tfn - fjac[(b*6 + k)*1024 + n]) * invdt[k];
    #pragma unroll
    for (int k = 0; k < 6; ++k) {
      float s = 0.f;
      #pragma unroll
      for (int j2 = 0; j2 < 6; ++j2) s += sHinv[k*6 + j2] * j6[j2];
      pinvM[(b*6 + k)*1024 + n] = s;
    }
  }
}

__global__ __launch_bounds__(256)
void k_init_est(float* __restrict__ estT) {
  const int tid = threadIdx.x;     // 256 = 16 batches x 16 entries
  const int ij = tid & 15;
  estT[tid] = (ij % 5 == 0) ? 1.f : 0.f;
}

__global__ __launch_bounds__(256)
void k_transform(const float* __restrict__ s0, const float* __restrict__ estT,
                 float* __restrict__ p, const int* maxiter, int iter) {
  if (iter >= *maxiter) return;
  const int b = blockIdx.x;
  const int tid = threadIdx.x;
  __shared__ float T[16];
  if (tid < 16) T[tid] = estT[b*16 + tid];
  __syncthreads();
  for (int n = tid; n < NPTS; n += 256) {
    const float* q = s0 + (b*NPTS + n)*3;
    const float X = q[0], Y = q[1], Z = q[2];
    float* o = p + (b*NPTS + n)*3;
    o[0] = T[0]*X + T[1]*Y + T[2] *Z + T[3];
    o[1] = T[4]*X + T[5]*Y + T[6] *Z + T[7];
    o[2] = T[8]*X + T[9]*Y + T[10]*Z + T[11];
  }
}

__global__ __launch_bounds__(256)
void k_pose(const float* __restrict__ sf, const float* __restrict__ tf,
            const float* __restrict__ pinvM, float* __restrict__ pose,
            const int* maxiter, int iter) {
  if (iter >= *maxiter) return;
  const int b = blockIdx.x;
  const int tid = threadIdx.x;
  __shared__ float sp[6];
  if (tid < 6) sp[tid] = 0.f;
  __syncthreads();
  float acc[6];
  #pragma unroll
  for (int k = 0; k < 6; ++k) acc[k] = 0.f;
  for (int n = tid; n < NPTS; n += 256) {
    const float r = sf[b*1024 + n] - tf[b*1024 + n];
    #pragma unroll
    for (int k = 0; k < 6; ++k) acc[k] += pinvM[(b*6 + k)*1024 + n] * r;
  }
  #pragma unroll
  for (int k = 0; k < 6; ++k) atomicAdd(&sp[k], acc[k]);
  __syncthreads();
  if (tid < 6) pose[b*6 + tid] = -sp[tid];
}

__global__ __launch_bounds__(64)
void k_apply(const float* __restrict__ pose, float* __restrict__ estT,
             const int* maxiter, int iter) {
  if (iter >= *maxiter) return;
  __shared__ float snorm[16];
  const int tid = threadIdx.x;
  if (tid < 16) {
    const float* p = pose + tid*6;
    float s = 1e-30f;
    #pragma unroll
    for (int i = 0; i < 6; ++i) s += p[i]*p[i];
    snorm[tid] = sqrtf(s);
  }
  __syncthreads();
  if (tid == 0) {
    float m = 0.f;
    for (int i = 0; i < 16; ++i) m = fmaxf(m, snorm[i]);
    snorm[0] = m;
  }
  __syncthreads();
  if (snorm[0] < XTOL_F) return;   // scalar check across all batches (reference)
  if (tid < 16) {
    float x6[6];
    #pragma unroll
    for (int i = 0; i < 6; ++i) x6[i] = pose[tid*6 + i];
    float G[16];
    exp_se3_dev(x6, G);
    float* T = estT + tid*16;
    float N[16];
    #pragma unroll
    for (int i = 0; i < 4; ++i)
      #pragma unroll
      for (int j = 0; j < 4; ++j)
        N[i*4+j] = G[i*4+0]*T[0*4+j] + G[i*4+1]*T[1*4+j]
                 + G[i*4+2]*T[2*4+j] + G[i*4+3]*T[3*4+j];
    #pragma unroll
    for (int i = 0; i < 16; ++i) T[i] = N[i];
  }
}

__global__ __launch_bounds__(64)
void k_final(const float* __restrict__ estT, const float* __restrict__ m0,
             const float* __restrict__ m1, float* __restrict__ out) {
  const int b = threadIdx.x;
  if (b >= 16) return;
  const float* T  = estT + b*16;
  const float* p0 = m0 + b*3;
  const float* p1 = m1 + b*3;
  #pragma unroll
  for (int i = 0; i < 4; ++i) {
    #pragma unroll
    for (int j = 0; j < 3; ++j) out[b*16 + i*4 + j] = T[i*4 + j];
    const float tcol = T[i*4+3] + ((i < 3) ? p0[i] : 0.f)
                     - (T[i*4+0]*p1[0] + T[i*4+1]*p1[1] + T[i*4+2]*p1[2]);
    out[b*16 + i*4 + 3] = tcol;
  }
}

// ---------------------------------------------------------------------------
// Launch
// ---------------------------------------------------------------------------
extern "C" void kernel_launch(void* const* d_in, const int* in_sizes, int n_in,
                              void* d_out, int out_size, void* d_ws, size_t ws_size,
                              hipStream_t stream) {
  const float* tpl = (const float*)d_in[0];
  const float* src = (const float*)d_in[1];
  const float* W1 = (const float*)d_in[2];  const float* b1 = (const float*)d_in[3];
  const float* W2 = (const float*)d_in[4];  const float* b2 = (const float*)d_in[5];
  const float* W3 = (const float*)d_in[6];  const float* b3 = (const float*)d_in[7];
  const float* W4 = (const float*)d_in[8];  const float* b4 = (const float*)d_in[9];
  const float* W5 = (const float*)d_in[10]; const float* b5 = (const float*)d_in[11];
  const float* dt = (const float*)d_in[12];
  const int* maxiter = (const int*)d_in[13];
  float* out = (float*)d_out;

  // Workspace layout (floats), total ~672K floats (~2.7 MB).
  float* ws = (float*)d_ws;
  float* p0m   = ws + 0;        // 48
  float* p1m   = ws + 48;       // 48
  float* t0    = ws + 96;       // 16*1024*3
  float* s0    = ws + 49248;    // 16*1024*3
  float* jp    = ws + 98400;    // 96*1024*3
  float* tf    = ws + 393312;   // 16*1024
  float* fjac  = ws + 409696;   // 96*1024
  float* pinvM = ws + 508000;   // 16*6*1024
  float* estT  = ws + 606304;   // 16*16
  float* pcur  = ws + 606560;   // 16*1024*3
  float* sf    = ws + 655712;   // 16*1024
  float* pose  = ws + 672096;   // 16*6

  k_center<<<32, 256, 0, stream>>>(tpl, src, p0m, p1m, t0, s0);
  k_jacpts<<<96, 256, 0, stream>>>(t0, dt, jp);

  // tf = feat(t0); fjac = feat(perturbed templates)   (no iteration guard)
  k_feat<<<16, 256, 0, stream>>>(t0, W1,b1,W2,b2,W3,b3,W4,b4,W5,b5, tf,
                                 (const int*)nullptr, 0);
  k_feat<<<96, 256, 0, stream>>>(jp, W1,b1,W2,b2,W3,b3,W4,b4,W5,b5, fjac,
                                 (const int*)nullptr, 0);
  k_pinv<<<16, 256, 0, stream>>>(tf, fjac, dt, pinvM);
  k_init_est<<<1, 256, 0, stream>>>(estT);

  // LK iterations: setup uses maxiter=10; launches are fixed for graph capture,
  // each kernel early-outs if iter >= *maxiter (device-side scalar).
  for (int it = 0; it < 10; ++it) {
    k_transform<<<16, 256, 0, stream>>>(s0, estT, pcur, maxiter, it);
    k_feat<<<16, 256, 0, stream>>>(pcur, W1,b1,W2,b2,W3,b3,W4,b4,W5,b5, sf,
                                   maxiter, it);
    k_pose<<<16, 256, 0, stream>>>(sf, tf, pinvM, pose, maxiter, it);
    k_apply<<<1, 64, 0, stream>>>(pose, estT, maxiter, it);
  }

  k_final<<<1, 64, 0, stream>>>(estT, p0m, p1m, out);
}